// CRF_80814104642178
// MI455X (gfx1250) — compile-verified
//
#include <hip/hip_runtime.h>

#define HIDDEN    1024
#define KTAGS     5
#define TLEN      32768
#define START_TAG 3
#define STOP_TAG  4
#define NEGV      -10000.0f
#define TCHUNK    128
#define NCHUNK    256      /* TLEN / TCHUNK */
#define ESTRIDE   8        /* padded emissions row stride (floats) */

typedef float v2f __attribute__((ext_vector_type(2)));
typedef float v8f __attribute__((ext_vector_type(8)));

// ---------------------------------------------------------------------------
// K1: emissions[t][k] = feats[t,:] . W[k,:] + b[k]   via V_WMMA_F32_16X16X4_F32
// One wave computes a 16-row tile; N padded 5->16 (cols 5..15 never stored).
// A frag (16x4 f32): lane<16 -> row=l15, K=k+0..1 ; lane>=16 -> row=l15, K=k+2..3
// B frag (4x16 f32): lane<16 -> N=l15, K=k+0..1 ; lane>=16 -> N=l15, K=k+2..3
// ---------------------------------------------------------------------------
__global__ __launch_bounds__(256) void k_emis(const float* __restrict__ feats,
                                              const float* __restrict__ W,
                                              const float* __restrict__ bias,
                                              float* __restrict__ emis) {
  const int lane = threadIdx.x & 31;
  const int wave = threadIdx.x >> 5;
  const int tile = blockIdx.x * 8 + wave;
  const int row0 = tile << 4;
  const int hi   = lane >> 4;          // which half-wave
  const int l15  = lane & 15;
  const int koff = hi << 1;            // 0 or 2

  const float* aptr = feats + (size_t)(row0 + l15) * HIDDEN + koff;
  const float* wptr = W + (size_t)(l15 % KTAGS) * HIDDEN + koff; // dup rows for N>=5

  v8f acc = {0.f, 0.f, 0.f, 0.f, 0.f, 0.f, 0.f, 0.f};
  for (int k0 = 0; k0 < HIDDEN; k0 += 64) {
    __builtin_prefetch(aptr + k0 + 256, 0, 1);   // global_prefetch_b8, 1KB ahead
#pragma unroll
    for (int kk = 0; kk < 64; kk += 4) {
      v2f a = *(const v2f*)(aptr + k0 + kk);
      v2f b = *(const v2f*)(wptr + k0 + kk);
      acc = __builtin_amdgcn_wmma_f32_16x16x4_f32(false, a, false, b,
                                                  (short)0, acc, false, false);
    }
  }
  // D layout: acc[v] of this lane = out[row0 + v + 8*hi][l15]
  if (l15 < KTAGS) {
    const float bv = bias[l15];
#pragma unroll
    for (int v = 0; v < 8; v++) {
      const int r = row0 + v + (hi << 3);
      emis[(size_t)r * ESTRIDE + l15] = acc[v] + bv;
    }
  }
}

// ---------------------------------------------------------------------------
// K2: per-chunk max-plus matrix product, column-parallel (5 lanes per chunk).
// Lane owns column p of P_c; step: col'[n] = max_q(trans[n][q]+col[q]) + e_t[n]
// ---------------------------------------------------------------------------
__global__ __launch_bounds__(256) void k_chunkmat(const float* __restrict__ emis,
                                                  const float* __restrict__ trans,
                                                  float* __restrict__ Pmat) {
  const int gid = blockIdx.x * 256 + threadIdx.x;   // 0..1279
  const int c = gid / KTAGS;
  const int p = gid % KTAGS;
  if (c >= NCHUNK) return;

  float tr[25];
#pragma unroll
  for (int i = 0; i < 25; i++) tr[i] = trans[i];

  float col[KTAGS];
#pragma unroll
  for (int q = 0; q < KTAGS; q++) col[q] = (q == p) ? 0.f : NEGV;  // identity column

  const float* e = emis + (size_t)c * TCHUNK * ESTRIDE;
  for (int t = 0; t < TCHUNK; t++) {
    float nc[KTAGS];
#pragma unroll
    for (int n = 0; n < KTAGS; n++) {
      float m = col[0] + tr[n * KTAGS + 0];
#pragma unroll
      for (int q = 1; q < KTAGS; q++) m = fmaxf(m, col[q] + tr[n * KTAGS + q]);
      nc[n] = m + e[t * ESTRIDE + n];
    }
#pragma unroll
    for (int n = 0; n < KTAGS; n++) col[n] = nc[n];
  }
#pragma unroll
  for (int n = 0; n < KTAGS; n++) Pmat[(size_t)c * 25 + n * KTAGS + p] = col[n];
}

// ---------------------------------------------------------------------------
// K3: serial scan over 256 chunk matrices; stores per-chunk entry vectors,
// final score (d_out[0]) and best terminal tag (first-max tie-break).
// ---------------------------------------------------------------------------
__global__ void k_scan(const float* __restrict__ Pmat, const float* __restrict__ trans,
                       float* __restrict__ entry_v, float* __restrict__ score_out,
                       int* __restrict__ best_out) {
  if (threadIdx.x != 0 || blockIdx.x != 0) return;
  float v[KTAGS];
#pragma unroll
  for (int n = 0; n < KTAGS; n++) v[n] = (n == START_TAG) ? 0.f : NEGV;

  for (int c = 0; c < NCHUNK; c++) {
#pragma unroll
    for (int n = 0; n < KTAGS; n++) entry_v[c * KTAGS + n] = v[n];
    const float* P = Pmat + (size_t)c * 25;
    float nv[KTAGS];
#pragma unroll
    for (int n = 0; n < KTAGS; n++) {
      float m = P[n * KTAGS + 0] + v[0];
#pragma unroll
      for (int p = 1; p < KTAGS; p++) m = fmaxf(m, P[n * KTAGS + p] + v[p]);
      nv[n] = m;
    }
#pragma unroll
    for (int n = 0; n < KTAGS; n++) v[n] = nv[n];
  }
  float best = -3.4e38f;
  int bi = 0;
#pragma unroll
  for (int n = 0; n < KTAGS; n++) {
    const float tv = v[n] + trans[STOP_TAG * KTAGS + n];
    if (tv > best) { best = tv; bi = n; }
  }
  score_out[0] = best;
  best_out[0] = bi;
}

// ---------------------------------------------------------------------------
// K4: per-chunk (1 lane each) vector recurrence from entry state; emits
// nibble-packed backpointers bp[t] and the chunk's composed backtrack map
// H_c = bp_{first} o ... o bp_{last} (also nibble-packed).
// ---------------------------------------------------------------------------
__global__ __launch_bounds__(256) void k_bp(const float* __restrict__ emis,
                                            const float* __restrict__ trans,
                                            const float* __restrict__ entry_v,
                                            unsigned* __restrict__ bp,
                                            unsigned* __restrict__ Htab) {
  const int c = blockIdx.x * 256 + threadIdx.x;
  if (c >= NCHUNK) return;

  float tr[25];
#pragma unroll
  for (int i = 0; i < 25; i++) tr[i] = trans[i];

  float v[KTAGS];
#pragma unroll
  for (int n = 0; n < KTAGS; n++) v[n] = entry_v[c * KTAGS + n];

  unsigned tbl = 0x43210u;  // identity map, nibble x -> x
  const float* e = emis + (size_t)c * TCHUNK * ESTRIDE;
  unsigned* bpc = bp + (size_t)c * TCHUNK;

  for (int t = 0; t < TCHUNK; t++) {
    unsigned packed = 0;
    float nv[KTAGS];
#pragma unroll
    for (int n = 0; n < KTAGS; n++) {
      float m = v[0] + tr[n * KTAGS + 0];
      int a = 0;
#pragma unroll
      for (int q = 1; q < KTAGS; q++) {
        const float x = v[q] + tr[n * KTAGS + q];
        if (x > m) { m = x; a = q; }   // strict > keeps first max (argmax semantics)
      }
      nv[n] = m + e[t * ESTRIDE + n];
      packed |= (unsigned)a << (4 * n);
    }
    bpc[t] = packed;
    // compose: tbl'[x] = tbl[ bp_t[x] ]  (all nibble shifts, no scratch arrays)
    unsigned nt = 0;
#pragma unroll
    for (int x = 0; x < KTAGS; x++) {
      const unsigned bx = (packed >> (4 * x)) & 0xFu;
      const unsigned tx = (tbl >> (4 * bx)) & 0xFu;
      nt |= tx << (4 * x);
    }
    tbl = nt;
#pragma unroll
    for (int n = 0; n < KTAGS; n++) v[n] = nv[n];
  }
  Htab[c] = tbl;
}

// ---------------------------------------------------------------------------
// K5: one 256-thread block. Thread 0 chains composed maps (in LDS) to get each
// chunk's entry carry; then all threads walk their chunk writing path[t].
// ---------------------------------------------------------------------------
__global__ __launch_bounds__(256) void k_backtrack(const unsigned* __restrict__ bp,
                                                   const unsigned* __restrict__ Htab,
                                                   const int* __restrict__ best_in,
                                                   float* __restrict__ path_out) {
  __shared__ unsigned sH[NCHUNK];
  __shared__ int sC[NCHUNK];
  const int c = threadIdx.x;
  sH[c] = Htab[c];
  __syncthreads();
  if (c == 0) {
    int carry = best_in[0];
    for (int cc = NCHUNK - 1; cc >= 0; cc--) {
      sC[cc] = carry;
      carry = (int)((sH[cc] >> (4 * carry)) & 0xFu);
    }
  }
  __syncthreads();
  int carry = sC[c];
  const unsigned* bpc = bp + (size_t)c * TCHUNK;
  for (int t = TCHUNK - 1; t >= 0; t--) {
    path_out[(size_t)c * TCHUNK + t] = (float)carry;   // path[t]
    carry = (int)((bpc[t] >> (4 * carry)) & 0xFu);     // bp_t[path[t]]
  }
}

// ---------------------------------------------------------------------------
extern "C" void kernel_launch(void* const* d_in, const int* in_sizes, int n_in,
                              void* d_out, int out_size, void* d_ws, size_t ws_size,
                              hipStream_t stream) {
  const float* feats = (const float*)d_in[0];   // [1, T, HIDDEN]
  const float* W     = (const float*)d_in[1];   // [K, HIDDEN]
  const float* b     = (const float*)d_in[2];   // [K]
  const float* trans = (const float*)d_in[3];   // [K, K]
  float* out = (float*)d_out;                   // [0]=score, [1..T]=path (as f32)

  char* ws = (char*)d_ws;
  size_t off = 0;
  auto alloc = [&](size_t bytes) -> char* {
    char* p = ws + off;
    off = (off + bytes + 255) & ~(size_t)255;
    return p;
  };
  float*    emis  = (float*)alloc((size_t)TLEN * ESTRIDE * sizeof(float));   // 1 MB
  float*    Pmat  = (float*)alloc((size_t)NCHUNK * 25 * sizeof(float));
  float*    entry = (float*)alloc((size_t)NCHUNK * KTAGS * sizeof(float));
  unsigned* bp    = (unsigned*)alloc((size_t)TLEN * sizeof(unsigned));       // 128 KB
  unsigned* Ht    = (unsigned*)alloc((size_t)NCHUNK * sizeof(unsigned));
  int*      bestp = (int*)alloc(sizeof(int));

  k_emis<<<TLEN / 16 / 8, 256, 0, stream>>>(feats, W, b, emis);
  k_chunkmat<<<(NCHUNK * KTAGS + 255) / 256, 256, 0, stream>>>(emis, trans, Pmat);
  k_scan<<<1, 32, 0, stream>>>(Pmat, trans, entry, out, bestp);
  k_bp<<<NCHUNK / 256, 256, 0, stream>>>(emis, trans, entry, bp, Ht);
  k_backtrack<<<1, NCHUNK, 0, stream>>>(bp, Ht, bestp, out + 1);
}